// Get_gradient_nopadding_msi_3487513444493
// MI455X (gfx1250) — compile-verified
//
#include <hip/hip_runtime.h>
#include <stdint.h>

// Sobel-like gradient magnitude, (16,31,256,256) fp32, zero padding = 1.
// Memory-bound (AI ~ 0.8 flop/byte): staged through LDS with the CDNA5
// Tensor Data Mover (tensor_load_to_lds, TENSORcnt), computed from LDS with
// b128 loads, written with b128 non-temporal stores.

typedef __attribute__((ext_vector_type(4))) float        v4f;
typedef __attribute__((ext_vector_type(4))) unsigned int v4u;
typedef __attribute__((ext_vector_type(4))) int          v4i_t;
typedef __attribute__((ext_vector_type(8))) int          v8i_t;

#define NC       496          // 16 * 31 images
#define H        256
#define W        256
#define TILE_H   16
#define LDS_ROWS (TILE_H + 2) // tile rows + top/bottom halo

__global__ __launch_bounds__(256)
void sobel_mag_kernel(const float* __restrict__ x, float* __restrict__ out) {
    __shared__ float tile[LDS_ROWS][W];   // 18 * 256 * 4 = 18432 B

    const int tid   = threadIdx.x;
    const int blk   = blockIdx.x;
    const int img   = blk >> 4;           // blk / (H/TILE_H)
    const int tiley = blk & 15;
    const int y0    = tiley * TILE_H;

    const uint32_t lds_base = (uint32_t)(uintptr_t)(&tile[0][0]);
    const size_t   img_base = (size_t)img * (size_t)(H * W);

    // ---- Valid global row range for this tile (clamped at image edges).
    const int gr_lo    = (y0 == 0) ? 0 : (y0 - 1);
    const int gr_hi    = (y0 + TILE_H < H) ? (y0 + TILE_H) : (H - 1);
    const int nrows    = gr_hi - gr_lo + 1;          // 17 or 18
    const int lds_row0 = (y0 == 0) ? 1 : 0;          // LDS row where DMA lands

    // ---- Zero-fill the one missing halo row at image top/bottom (zero pad).
    if (y0 == 0 && tid < 64) {
        v4f z = {0.f, 0.f, 0.f, 0.f};
        *(v4f*)&tile[0][tid << 2] = z;
    }
    if (y0 + TILE_H == H && tid < 64) {
        v4f z = {0.f, 0.f, 0.f, 0.f};
        *(v4f*)&tile[LDS_ROWS - 1][tid << 2] = z;
    }

    // ---- One TDM DMA per block: 2D tile (nrows x 256 fp32) global -> LDS.
    // Descriptor layout per CDNA5 ISA ch.8 (D# groups 0/1).
    if (tid < 32) {   // single wave issues the DMA (EXEC is ignored by TDM)
        const uint64_t ga = (uint64_t)(uintptr_t)(x + img_base + (size_t)gr_lo * W);
        const uint32_t la = lds_base + (uint32_t)(lds_row0 * W * 4);

        const v4u g0 = {
            1u,                                               // count=1, user D#
            la,                                               // lds_addr (bytes)
            (uint32_t)ga,                                     // global_addr[31:0]
            ((uint32_t)(ga >> 32) & 0x01FFFFFFu) | 0x80000000u // ga[56:32] | type=2
        };
        const v8i_t g1 = {
            0x00020000,            // wg_mask=0, data_size=2 (4B), no pad/iter
            (int)(W << 16),        // tensor_dim0 = 256 (bits 79:48, low part)
            (int)(nrows << 16),    // tensor_dim0 hi=0 | tensor_dim1 lo = nrows
            (int)(W << 16),        // tensor_dim1 hi=0 | tile_dim0 = 256
            (int)nrows,            // tile_dim1 = nrows | tile_dim2 = 0
            (int)W,                // tensor_dim0_stride = 256 (lo 32)
            0,                     // dim0_stride hi | dim1_stride lo (unused, 2D)
            0                      // dim1_stride hi (unused)
        };
        const v4i_t z4 = {0, 0, 0, 0};

#if __has_include(<hip/amd_detail/amd_gfx1250_TDM.h>)
        const v8i_t z8 = {0, 0, 0, 0, 0, 0, 0, 0};
        __builtin_amdgcn_tensor_load_to_lds(g0, g1, z4, z4, z8, 0);   // 6-arg lane
#else
        __builtin_amdgcn_tensor_load_to_lds(g0, g1, z4, z4, 0);       // ROCm 7.2
#endif
        __builtin_amdgcn_s_wait_tensorcnt(0);   // issuing wave's DMA landed
    }
    __syncthreads();                             // visible to all 8 waves

    // ---- Compute 16 rows x 256 cols from LDS; 4 x float4 per thread.
    const int cg = tid & 63;        // column group: x0 = 4*cg
    const int r0 = tid >> 6;        // 0..3
    const int x0 = cg << 2;

#pragma unroll
    for (int i = 0; i < 4; ++i) {
        const int r = r0 + (i << 2);                  // output row in tile 0..15
        // LDS rows: up = r, mid = r+1, dn = r+2  (mid == global row y0+r)
        const v4f up  = *(const v4f*)&tile[r    ][x0];
        const v4f dn  = *(const v4f*)&tile[r + 2][x0];
        const v4f mid = *(const v4f*)&tile[r + 1][x0];
        const float left  = (x0 == 0)     ? 0.0f : tile[r + 1][x0 - 1];
        const float right = (x0 + 4 >= W) ? 0.0f : tile[r + 1][x0 + 4];

        v4f v = dn - up;
        v4f h;
        h.x = mid.y - left;
        h.y = mid.z - mid.x;
        h.z = mid.w - mid.y;
        h.w = right - mid.z;

        v4f m = v * v + h * h + 1e-6f;
        v4f res;
        res.x = __builtin_sqrtf(m.x);
        res.y = __builtin_sqrtf(m.y);
        res.z = __builtin_sqrtf(m.z);
        res.w = __builtin_sqrtf(m.w);

        // Output is never re-read: non-temporal b128 store keeps the 130 MB
        // result stream from evicting input halo lines in the 192 MB L2.
        float* op = out + img_base + (size_t)(y0 + r) * W + (size_t)x0;
        __builtin_nontemporal_store(res, (v4f*)op);
    }
}

extern "C" void kernel_launch(void* const* d_in, const int* in_sizes, int n_in,
                              void* d_out, int out_size, void* d_ws, size_t ws_size,
                              hipStream_t stream) {
    const float* x   = (const float*)d_in[0];
    float*       out = (float*)d_out;

    dim3 grid(NC * (H / TILE_H));   // 496 * 16 = 7936 blocks
    dim3 block(256);                // 8 wave32 waves
    sobel_mag_kernel<<<grid, block, 0, stream>>>(x, out);
}